// PixelDSNTDistanceDoublePredict_55035710931300
// MI455X (gfx1250) — compile-verified
//
#include <hip/hip_runtime.h>
#include <stdint.h>

#define Bn 32
#define Cn 2
#define Hn 512
#define Wn 512
#define NH (Bn*Cn)            // 64 heatmaps
#define HW (Hn*Wn)            // 262144 elements per heatmap
#define PARTS 32              // blocks per heatmap
#define CHUNK (HW/PARTS)      // 8192 elements per block
#define TPB 256               // 8 wave32 per block

typedef float v2f __attribute__((ext_vector_type(2)));
typedef float v8f __attribute__((ext_vector_type(8)));

// Monotonic unsigned key for float ordering (NaN-free inputs)
__device__ __forceinline__ unsigned ordKey(float f) {
    unsigned u = __float_as_uint(f);
    return (u & 0x80000000u) ? ~u : (u | 0x80000000u);
}

// Full 32-lane sum using one V_WMMA_F32_16X16X4_F32:
//   A[m][0] = v_m (lanes 0-15, VGPR0), A[m][2] = v_{m+16} (lanes 16-31, VGPR0),
//   A[*][1] = A[*][3] = 0, B = ones  =>  D[m][n] = v_m + v_{m+16}.
// Lane L<16 holds rows 0..7 in c[0..7]; lanes 16-31 hold rows 8..15.
// Sum of the 8 accumulators + shfl_xor(16) = full wave sum in every lane.
__device__ __forceinline__ float wmmaWaveSum(float v) {
    v2f a; a[0] = v;   a[1] = 0.0f;
    v2f b; b[0] = 1.0f; b[1] = 1.0f;
    v8f c = {};
    c = __builtin_amdgcn_wmma_f32_16x16x4_f32(false, a, false, b,
                                              (short)0, c, false, false);
    float r = ((c[0] + c[1]) + (c[2] + c[3])) + ((c[4] + c[5]) + (c[6] + c[7]));
    r += __shfl_xor(r, 16, 32);
    return r;
}

__device__ __forceinline__ unsigned long long shflxor64(unsigned long long x, int m) {
    unsigned lo = (unsigned)x, hi = (unsigned)(x >> 32);
    lo = __shfl_xor(lo, m, 32);
    hi = __shfl_xor(hi, m, 32);
    return ((unsigned long long)hi << 32) | lo;
}

// One block = one 8192-element chunk of one heatmap.
// Single read of input (online softmax) + single read of target (packed argmax).
__global__ __launch_bounds__(TPB) void dsnt_partials(
    const float* __restrict__ inp, const float* __restrict__ tgt,
    float* __restrict__ Pm, float* __restrict__ Ps,
    float* __restrict__ Psx, float* __restrict__ Psy,
    unsigned long long* __restrict__ Ppk)
{
    const int blk   = blockIdx.x;
    const int hm    = blk / PARTS;
    const int part  = blk % PARTS;
    const int tid   = threadIdx.x;
    const int lane  = tid & 31;
    const int wave  = tid >> 5;
    const int cbase = part * CHUNK;

    const float4* ip = (const float4*)(inp + (size_t)hm * HW + cbase);
    const float4* tp = (const float4*)(tgt + (size_t)hm * HW + cbase);

    // ---- input pass: b128 loads, thread-local max first (1 exp/element later)
    float4 q[8];
    float m = -3.402823466e38f;
    #pragma unroll
    for (int i = 0; i < 8; ++i) {
        q[i] = ip[tid + i * TPB];
        m = fmaxf(m, fmaxf(fmaxf(q[i].x, q[i].y), fmaxf(q[i].z, q[i].w)));
    }

    float s = 0.0f, sx = 0.0f, sy = 0.0f;
    #pragma unroll
    for (int i = 0; i < 8; ++i) {
        const int off = cbase + (tid + i * TPB) * 4;     // flat idx in heatmap
        const float fy = (float)((off >> 9) + 1);        // row weight (h+1)
        const float fx = (float)((off & 511) + 1);       // col weight (w+1)
        const float e0 = __expf(q[i].x - m);
        const float e1 = __expf(q[i].y - m);
        const float e2 = __expf(q[i].z - m);
        const float e3 = __expf(q[i].w - m);
        const float es = (e0 + e1) + (e2 + e3);
        s  += es;
        sy += es * fy;
        sx += e0 * fx + e1 * (fx + 1.0f) + e2 * (fx + 2.0f) + e3 * (fx + 3.0f);
    }

    // wave max, rescale once per lane, then WMMA 32-lane sums
    float mw = m;
    #pragma unroll
    for (int d = 1; d < 32; d <<= 1) mw = fmaxf(mw, __shfl_xor(mw, d, 32));
    const float sc = __expf(m - mw);
    s  = wmmaWaveSum(s  * sc);
    sx = wmmaWaveSum(sx * sc);
    sy = wmmaWaveSum(sy * sc);

    // ---- target pass: packed (ordKey<<32)|(~idx) => max = argmax, first-occurrence
    unsigned long long pk = 0ull;
    #pragma unroll
    for (int i = 0; i < 8; ++i) {
        const float4 t = tp[tid + i * TPB];
        const unsigned off = (unsigned)(cbase + (tid + i * TPB) * 4);
        unsigned long long p;
        p = ((unsigned long long)ordKey(t.x) << 32) | (0xFFFFFFFFu - (off + 0u)); if (p > pk) pk = p;
        p = ((unsigned long long)ordKey(t.y) << 32) | (0xFFFFFFFFu - (off + 1u)); if (p > pk) pk = p;
        p = ((unsigned long long)ordKey(t.z) << 32) | (0xFFFFFFFFu - (off + 2u)); if (p > pk) pk = p;
        p = ((unsigned long long)ordKey(t.w) << 32) | (0xFFFFFFFFu - (off + 3u)); if (p > pk) pk = p;
    }
    #pragma unroll
    for (int d = 1; d < 32; d <<= 1) {
        unsigned long long o = shflxor64(pk, d);
        if (o > pk) pk = o;
    }

    // ---- cross-wave combine in LDS, thread 0 writes this block's partial slot
    __shared__ float lm[8], ls[8], lsx[8], lsy[8];
    __shared__ unsigned long long lpk[8];
    if (lane == 0) { lm[wave] = mw; ls[wave] = s; lsx[wave] = sx; lsy[wave] = sy; lpk[wave] = pk; }
    __syncthreads();
    if (tid == 0) {
        float M = lm[0], S = ls[0], SX = lsx[0], SY = lsy[0];
        unsigned long long PK = lpk[0];
        #pragma unroll
        for (int i = 1; i < 8; ++i) {
            const float mi = lm[i];
            const float nM = fmaxf(M, mi);
            const float a0 = __expf(M - nM);
            const float a1 = __expf(mi - nM);
            S  = S  * a0 + ls[i]  * a1;
            SX = SX * a0 + lsx[i] * a1;
            SY = SY * a0 + lsy[i] * a1;
            M = nM;
            if (lpk[i] > PK) PK = lpk[i];
        }
        Pm[blk] = M; Ps[blk] = S; Psx[blk] = SX; Psy[blk] = SY; Ppk[blk] = PK;
    }
}

// One block, 64 threads: fold 32 partials per heatmap, emit all outputs.
__global__ __launch_bounds__(64) void dsnt_finalize(
    const float* __restrict__ Pm, const float* __restrict__ Ps,
    const float* __restrict__ Psx, const float* __restrict__ Psy,
    const unsigned long long* __restrict__ Ppk, float* __restrict__ out)
{
    const int hm = threadIdx.x;   // heatmap = b*C + c
    __shared__ float spx[NH], spy[NH], stx[NH], sty[NH], sed[NH];

    if (hm < NH) {
        float M = -3.402823466e38f, S = 0.0f, SX = 0.0f, SY = 0.0f;
        unsigned long long PK = 0ull;
        for (int i = 0; i < PARTS; ++i) {
            const int idx = hm * PARTS + i;
            const float mi = Pm[idx];
            const float nM = fmaxf(M, mi);
            const float a0 = __expf(M - nM);
            const float a1 = __expf(mi - nM);
            S  = S  * a0 + Ps[idx]  * a1;
            SX = SX * a0 + Psx[idx] * a1;
            SY = SY * a0 + Psy[idx] * a1;
            M = nM;
            const unsigned long long p = Ppk[idx];
            if (p > PK) PK = p;
        }
        const float px = SX / S;
        const float py = SY / S;
        const unsigned uidx = 0xFFFFFFFFu - (unsigned)(PK & 0xFFFFFFFFull);
        const float tx = (float)((uidx & 511u) + 1u);
        const float ty = (float)((uidx >> 9) + 1u);
        const float dx = tx - px, dy = ty - py;
        const float ed = sqrtf(dx * dx + dy * dy);

        out[4 + hm * 2 + 0] = px - 1.0f;             // pred_coords
        out[4 + hm * 2 + 1] = py - 1.0f;
        out[4 + 2 * NH + hm * 2 + 0] = tx - 1.0f;    // true_coords
        out[4 + 2 * NH + hm * 2 + 1] = ty - 1.0f;
        spx[hm] = px; spy[hm] = py; stx[hm] = tx; sty[hm] = ty; sed[hm] = ed;
    }
    __syncthreads();
    if (hm == 0) {
        float si = 0.0f, ss = 0.0f, dd = 0.0f;
        for (int b = 0; b < Bn; ++b) {
            si += sed[2 * b];
            ss += sed[2 * b + 1];
            const float pdx = spx[2 * b] - spx[2 * b + 1];
            const float pdy = spy[2 * b] - spy[2 * b + 1];
            const float tdx = stx[2 * b] - stx[2 * b + 1];
            const float tdy = sty[2 * b] - sty[2 * b + 1];
            dd += sqrtf(pdx * pdx + pdy * pdy) - sqrtf(tdx * tdx + tdy * tdy);
        }
        const float inv = 1.0f / (float)Bn;
        out[0] = si * inv;
        out[1] = ss * inv;
        out[2] = (si + ss) * inv;
        out[3] = dd * inv;
    }
}

extern "C" void kernel_launch(void* const* d_in, const int* in_sizes, int n_in,
                              void* d_out, int out_size, void* d_ws, size_t ws_size,
                              hipStream_t stream)
{
    const float* inp = (const float*)d_in[0];
    const float* tgt = (const float*)d_in[1];

    char* ws = (char*)d_ws;                                   // 48 KiB used
    unsigned long long* Ppk = (unsigned long long*)(ws);      // 2048 * 8  = 16384 B
    float* Pm  = (float*)(ws + 16384);                        // 2048 * 4
    float* Ps  = (float*)(ws + 24576);
    float* Psx = (float*)(ws + 32768);
    float* Psy = (float*)(ws + 40960);

    dsnt_partials<<<NH * PARTS, TPB, 0, stream>>>(inp, tgt, Pm, Ps, Psx, Psy, Ppk);
    dsnt_finalize<<<1, 64, 0, stream>>>(Pm, Ps, Psx, Psy, Ppk, (float*)d_out);
}